// GlobalNormFlipFlop_7662221656150
// MI455X (gfx1250) — compile-verified
//
#include <hip/hip_runtime.h>
#include <math.h>

// ---------------- problem constants (match reference setup_inputs) ----------
#define T_STEPS   2048
#define NBATCH    128
#define INSIZE    256
#define NSTATE    40          // 2*NBASE*(NBASE+1)
#define NROWS     (T_STEPS * NBATCH)       // 262144 GEMM rows
#define OUT_TOTAL (NROWS * NSTATE)         // 10,485,760 floats

// GEMM tiling
#define MT_PER_BLK  8
#define LDSX_STRIDE 260                    // 256 + 4 pad floats (16B-aligned rows, conflict-free)
#define LDSW_STRIDE 258                    // 256 + 2 pad floats (8B-aligned rows, conflict-free)
#define WROWS       41                     // 40 real W rows + 1 zero row for padded cols

// scan chunking
#define P_CHUNKS   64
#define CHUNK_LEN  (T_STEPS / P_CHUNKS)    // 32
#define NEG_BIG    (-1.0e30f)
#define LOG2E_F    1.44269504088896340736f
#define LN2_F      0.69314718055994530942f

typedef float v2f __attribute__((ext_vector_type(2)));
typedef float v8f __attribute__((ext_vector_type(8)));

// ---------------------------------------------------------------------------
// Fast transcendental wrappers: force single-instruction gfx1250 paths.
// Device pass uses AMDGCN builtins (v_exp_f32 / v_log_f32 / v_tanh_f32);
// host pass only needs these to be semantically valid.
// ---------------------------------------------------------------------------
__device__ __forceinline__ float fast_exp2(float v) {
#if defined(__AMDGCN__) && __has_builtin(__builtin_amdgcn_exp2f)
    return __builtin_amdgcn_exp2f(v);          // v_exp_f32 (2^x)
#else
    return exp2f(v);
#endif
}

__device__ __forceinline__ float fast_log2(float v) {
#if defined(__AMDGCN__) && __has_builtin(__builtin_amdgcn_logf)
    return __builtin_amdgcn_logf(v);           // v_log_f32 (log2 x)
#else
    return log2f(v);
#endif
}

__device__ __forceinline__ float fast_tanh(float v) {
#if defined(__AMDGCN__)
#if __has_builtin(__builtin_amdgcn_tanhf)
    return __builtin_amdgcn_tanhf(v);          // v_tanh_f32
#elif __has_builtin(__builtin_amdgcn_tanh_f32)
    return __builtin_amdgcn_tanh_f32(v);
#else
    float r;
    // v_nop covers the TRANS32 result-use hazard the compiler can't see.
    asm volatile("v_tanh_f32 %0, %1\n\tv_nop" : "=v"(r) : "v"(v));
    return r;
#endif
#else
    return tanhf(v);
#endif
}

// ---------------------------------------------------------------------------
// Kernel A: y = 5*tanh(x * W^T + b), f32 WMMA 16x16x4.
// Block = 96 threads (3 waves); wave w owns column tile w (cols 16w..16w+15).
// Each block processes MT_PER_BLK row tiles of 16, staging x through LDS and
// keeping W (plus one zero row) resident in LDS for the whole block.
// ---------------------------------------------------------------------------
__global__ __launch_bounds__(96) void gemm_tanh_kernel(
    const float* __restrict__ x, const float* __restrict__ W,
    const float* __restrict__ b, float* __restrict__ y)
{
    __shared__ float ldsW[WROWS * LDSW_STRIDE];
    __shared__ float ldsX[16 * LDSX_STRIDE];

    const int tid  = threadIdx.x;
    const int wave = tid >> 5;      // 0..2 -> column tile
    const int lane = tid & 31;

    // ---- stage W (40x256) into LDS; row 40 = zeros for the padded columns --
    for (int i = tid; i < 40 * 128; i += 96) {          // 128 float2 per row
        const int row = i >> 7;
        const int kk  = (i & 127) << 1;
        *reinterpret_cast<float2*>(&ldsW[row * LDSW_STRIDE + kk]) =
            *reinterpret_cast<const float2*>(W + row * INSIZE + kk);
    }
    for (int i = tid; i < 256; i += 96) ldsW[40 * LDSW_STRIDE + i] = 0.0f;

    // ---- per-lane fragment geometry (V_WMMA_F32_16X16X4_F32) ---------------
    const int rowA = lane & 15;                 // A: M index
    const int kOff = (lane >> 4) << 1;          // A/B: K pair 0..1 or 2..3
    const int colL = wave * 16 + (lane & 15);   // global output column 0..47
    const int wrow = (colL < NSTATE) ? colL : 40;   // clamp pad cols to zero row

    const float* pA = &ldsX[rowA * LDSX_STRIDE + kOff];
    const float* pB = &ldsW[wrow * LDSW_STRIDE + kOff];

    for (int mt = 0; mt < MT_PER_BLK; ++mt) {
        const long mtile = (long)blockIdx.x * MT_PER_BLK + mt;
        const float* xt = x + mtile * 16 * INSIZE;

        __syncthreads();                        // W staged / previous tile done
        for (int i = tid; i < 1024; i += 96) {  // 16 rows x 64 float4
            const int row = i >> 6;
            const int k4  = (i & 63) << 2;
            *reinterpret_cast<float4*>(&ldsX[row * LDSX_STRIDE + k4]) =
                *reinterpret_cast<const float4*>(xt + row * INSIZE + k4);
        }
        __syncthreads();

        v8f c = {};                             // f32 accumulator, 8 VGPRs
#pragma unroll 8
        for (int k0 = 0; k0 < INSIZE; k0 += 4) {
            v2f a  = *reinterpret_cast<const v2f*>(pA + k0);   // ds_load_b64
            v2f bb = *reinterpret_cast<const v2f*>(pB + k0);   // ds_load_b64
            c = __builtin_amdgcn_wmma_f32_16x16x4_f32(
                    false, a, false, bb, (short)0, c, false, false);
        }

        if (colL < NSTATE) {
            const float bias = b[colL];
            const long  rbase = mtile * 16 + ((lane >> 4) << 3); // +8 for hi half
#pragma unroll
            for (int v = 0; v < 8; ++v) {
                const float val = 5.0f * fast_tanh(c[v] + bias);
                y[(rbase + v) * (long)NSTATE + colL] = val;
            }
        }
    }
}

// ---------------------------------------------------------------------------
// log2-domain log-sum-exp helpers (bare v_exp_f32 / v_log_f32, no rescaling)
// ---------------------------------------------------------------------------
__device__ __forceinline__ float lse8_2(const float* a) {
    float m = a[0];
#pragma unroll
    for (int i = 1; i < 8; ++i) m = fmaxf(m, a[i]);
    float s = 0.0f;
#pragma unroll
    for (int i = 0; i < 8; ++i) s += fast_exp2(a[i] - m);
    return m + fast_log2(s);
}

// load 40 scores and convert natural-log domain -> log2 domain
__device__ __forceinline__ void load40_log2(float* sc, const float* p) {
#pragma unroll
    for (int i = 0; i < 10; ++i) {
        const float4 v = reinterpret_cast<const float4*>(p)[i];
        sc[i * 4 + 0] = v.x * LOG2E_F; sc[i * 4 + 1] = v.y * LOG2E_F;
        sc[i * 4 + 2] = v.z * LOG2E_F; sc[i * 4 + 3] = v.w * LOG2E_F;
    }
}

// ---------------------------------------------------------------------------
// Kernel B: per (chunk p, batch n) compose CHUNK_LEN sparse flip-flop
// transition matrices into a dense 8x8 matrix in the (log2,+) semiring.
// One thread per (p, n): grid = P_CHUNKS, block = NBATCH.
// Column s of M' depends only on column s of M -> column-wise in-place update.
// ---------------------------------------------------------------------------
__global__ __launch_bounds__(NBATCH) void chunk_mats_kernel(
    const float* __restrict__ y, float* __restrict__ mats)
{
    const int p = blockIdx.x;
    const int n = threadIdx.x;
    const float* base = y + (long)p * CHUNK_LEN * (NBATCH * NSTATE) + n * NSTATE;

    float sc[NSTATE];
    float M[8][8];

    // step 0: M = transition matrix itself (log2 domain)
    load40_log2(sc, base);
#pragma unroll
    for (int d = 0; d < 4; ++d)
#pragma unroll
        for (int s = 0; s < 8; ++s) M[d][s] = sc[d * 8 + s];
#pragma unroll
    for (int i = 0; i < 4; ++i) {
#pragma unroll
        for (int s = 0; s < 8; ++s) M[4 + i][s] = NEG_BIG;
        M[4 + i][i]     = sc[32 + i];
        M[4 + i][4 + i] = sc[36 + i];
    }

    for (int t = 1; t < CHUNK_LEN; ++t) {
        const float* pt = base + (long)t * (NBATCH * NSTATE);
        load40_log2(sc, pt);
        if (t + 1 < CHUNK_LEN)
            __builtin_prefetch(base + (long)(t + 1) * (NBATCH * NSTATE), 0, 1);

#pragma unroll
        for (int s = 0; s < 8; ++s) {
            float col[8];
#pragma unroll
            for (int k = 0; k < 8; ++k) col[k] = M[k][s];
            float nw[8];
#pragma unroll
            for (int d = 0; d < 4; ++d) {               // flip rows: dense lse8
                float a[8];
#pragma unroll
                for (int k = 0; k < 8; ++k) a[k] = sc[d * 8 + k] + col[k];
                nw[d] = lse8_2(a);
            }
#pragma unroll
            for (int i = 0; i < 4; ++i) {               // flop rows: 2-term lae
                const float av = sc[32 + i] + col[i];
                const float bv = sc[36 + i] + col[4 + i];
                const float mx = fmaxf(av, bv);
                nw[4 + i] = mx + fast_log2(fast_exp2(av - mx) + fast_exp2(bv - mx));
            }
#pragma unroll
            for (int k = 0; k < 8; ++k) M[k][s] = nw[k];
        }
    }

    float* out = mats + ((long)p * NBATCH + n) * 64;
#pragma unroll
    for (int d = 0; d < 8; ++d)
#pragma unroll
        for (int s = 0; s < 8; ++s) out[d * 8 + s] = M[d][s];
}

// ---------------------------------------------------------------------------
// Kernel C: fold the 64 chunk matrices per batch element, apply the initial
// state (4 flip states at 0, flop states exp(-50000)=0), emit logZ/T in
// natural-log domain (single x ln2 conversion at the end).
// ---------------------------------------------------------------------------
__global__ __launch_bounds__(NBATCH) void combine_logz_kernel(
    const float* __restrict__ mats, float* __restrict__ logZcorr)
{
    const int n = threadIdx.x;
    float g[8];

    const float* M0 = mats + (long)n * 64;              // chunk 0 applied to g0
#pragma unroll
    for (int d = 0; d < 8; ++d) {
        const float a0 = M0[d * 8 + 0], a1 = M0[d * 8 + 1];
        const float a2 = M0[d * 8 + 2], a3 = M0[d * 8 + 3];
        const float m = fmaxf(fmaxf(a0, a1), fmaxf(a2, a3));
        g[d] = m + fast_log2(fast_exp2(a0 - m) + fast_exp2(a1 - m) +
                             fast_exp2(a2 - m) + fast_exp2(a3 - m));
    }

    for (int p = 1; p < P_CHUNKS; ++p) {
        const float* Mp = mats + ((long)p * NBATCH + n) * 64;
        float Ml[64];
#pragma unroll
        for (int i = 0; i < 16; ++i) {
            const float4 v = reinterpret_cast<const float4*>(Mp)[i];
            Ml[i * 4 + 0] = v.x; Ml[i * 4 + 1] = v.y;
            Ml[i * 4 + 2] = v.z; Ml[i * 4 + 3] = v.w;
        }
        float gn[8];
#pragma unroll
        for (int d = 0; d < 8; ++d) {
            float a[8];
#pragma unroll
            for (int k = 0; k < 8; ++k) a[k] = Ml[d * 8 + k] + g[k];
            gn[d] = lse8_2(a);
        }
#pragma unroll
        for (int k = 0; k < 8; ++k) g[k] = gn[k];
    }

    // back to natural log, pre-divided by T
    logZcorr[n] = lse8_2(g) * (LN2_F / (float)T_STEPS);
}

// ---------------------------------------------------------------------------
// Kernel D: out[t,n,c] -= logZ[n]/T   (pure bandwidth pass)
// ---------------------------------------------------------------------------
__global__ void sub_kernel(float* __restrict__ out,
                           const float* __restrict__ logZcorr, long total)
{
    const long i = (long)blockIdx.x * blockDim.x + threadIdx.x;
    if (i < total) {
        const int n = (int)((i / NSTATE) % NBATCH);
        out[i] -= logZcorr[n];
    }
}

// ---------------------------------------------------------------------------
extern "C" void kernel_launch(void* const* d_in, const int* in_sizes, int n_in,
                              void* d_out, int out_size, void* d_ws, size_t ws_size,
                              hipStream_t stream)
{
    const float* x = (const float*)d_in[0];   // (T, N, INSIZE) f32
    const float* W = (const float*)d_in[1];   // (NSTATE, INSIZE) f32
    const float* b = (const float*)d_in[2];   // (NSTATE,) f32
    float* y = (float*)d_out;                 // (T, N, NSTATE) f32

    float* mats  = (float*)d_ws;                              // P*N*64 floats (2 MB)
    float* logZc = mats + (size_t)P_CHUNKS * NBATCH * 64;     // N floats

    const int gemm_blocks = NROWS / (16 * MT_PER_BLK);        // 2048
    gemm_tanh_kernel<<<gemm_blocks, 96, 0, stream>>>(x, W, b, y);

    chunk_mats_kernel<<<P_CHUNKS, NBATCH, 0, stream>>>(y, mats);
    combine_logz_kernel<<<1, NBATCH, 0, stream>>>(mats, logZc);

    const long total = (long)OUT_TOTAL;
    const int  blocks = (int)((total + 255) / 256);
    sub_kernel<<<blocks, 256, 0, stream>>>(y, logZc, total);
}